// PNULoss_37366215475814
// MI455X (gfx1250) — compile-verified
//
#include <hip/hip_runtime.h>

// ---------------------------------------------------------------------------
// PNU loss, MI455X (gfx1250, wave32).
// Memory-bound streaming reduction: 134 MB @ 23.3 TB/s ~ 5.8 us floor.
// Wave-level reduction uses V_WMMA_F32_16X16X4_F32 with a ones A-matrix:
// D[m][n] = sum_k B[k][n]  (exact f32, independent of B's K->VGPR mapping).
// ---------------------------------------------------------------------------

typedef float v2f __attribute__((ext_vector_type(2)));
typedef float v8f __attribute__((ext_vector_type(8)));

#define PNU_ETA 0.5f

__global__ void pnu_init_ws(float* __restrict__ ws) {
    if (threadIdx.x < 8) ws[threadIdx.x] = 0.0f;
}

__global__ __launch_bounds__(256) void pnu_main(const float* __restrict__ y,
                                                const int* __restrict__ t,
                                                float* __restrict__ ws,
                                                int n) {
    // Per-thread accumulators:
    // a0 = sum idx_p * sigmoid(-y)   a1 = sum idx_p * sigmoid(y)
    // a2 = sum idx_n * sigmoid(y)    a3 = sum idx_u * sigmoid(y)
    // a4 = n_p   a5 = n_n   a6 = n_u   a7 = pad (0)
    float a0 = 0.f, a1 = 0.f, a2 = 0.f, a3 = 0.f;
    float a4 = 0.f, a5 = 0.f, a6 = 0.f, a7 = 0.f;

    const int gid    = blockIdx.x * blockDim.x + threadIdx.x;
    const int stride = gridDim.x * blockDim.x;
    const int nvec   = n >> 2;

    const float4* __restrict__ y4 = (const float4*)y;
    const int4*   __restrict__ t4 = (const int4*)t;

    auto accum = [&](float yy, int tt) {
        float s  = 1.0f / (1.0f + __expf(-yy));   // sigmoid(y)
        float fp = (tt == 1)  ? 1.0f : 0.0f;
        float fn = (tt == -1) ? 1.0f : 0.0f;
        float fu = (tt == 0)  ? 1.0f : 0.0f;
        a0 = fmaf(fp, 1.0f - s, a0);
        a1 = fmaf(fp, s, a1);
        a2 = fmaf(fn, s, a2);
        a3 = fmaf(fu, s, a3);
        a4 += fp;
        a5 += fn;
        a6 += fu;
    };

    for (int i = gid; i < nvec; i += stride) {
        float4 yv = y4[i];   // global_load_b128
        int4   tv = t4[i];   // global_load_b128
        accum(yv.x, tv.x);
        accum(yv.y, tv.y);
        accum(yv.z, tv.z);
        accum(yv.w, tv.w);
    }
    // Tail (N % 4 != 0); for N = 2^24 this is skipped entirely.
    int rem = n & 3;
    if (gid < rem) {
        int i = (nvec << 2) + gid;
        accum(y[i], t[i]);
    }

    // ---- wave32 reduction ------------------------------------------------
    const int lane = threadIdx.x & 31;

    // 3-stage XOR butterfly: every lane in each 8-lane group holds the
    // group sum of each accumulator.
    #pragma unroll
    for (int d = 1; d <= 4; d <<= 1) {
        a0 += __shfl_xor(a0, d, 32);
        a1 += __shfl_xor(a1, d, 32);
        a2 += __shfl_xor(a2, d, 32);
        a3 += __shfl_xor(a3, d, 32);
        a4 += __shfl_xor(a4, d, 32);
        a5 += __shfl_xor(a5, d, 32);
        a6 += __shfl_xor(a6, d, 32);
    }

    // Lane L contributes group-sum of accumulator (L & 7).
    int   r = lane & 7;
    float v = a0;
    v = (r == 1) ? a1 : v;
    v = (r == 2) ? a2 : v;
    v = (r == 3) ? a3 : v;
    v = (r == 4) ? a4 : v;
    v = (r == 5) ? a5 : v;
    v = (r == 6) ? a6 : v;
    v = (r == 7) ? a7 : v;

    // WMMA ones-trick: A = ones(16x4), B = v in slot 0 (4x16).
    // Column n of D = v[lane n] + v[lane n+16] (exact f32, K-mapping-free).
    v2f ones; ones[0] = 1.0f; ones[1] = 1.0f;
    v2f bm;   bm[0]   = v;    bm[1]   = 0.0f;
    v8f c = {};
    v8f dres = __builtin_amdgcn_wmma_f32_16x16x4_f32(
        false, ones, false, bm, (short)0, c, false, false);
    float s2  = dres[0];                       // lane n -> column n of row 0/8
    float tot = s2 + __shfl_xor(s2, 8, 32);    // lanes with lane&7==j: total[j]

    // ---- cross-wave (8 waves @ 256 threads) ------------------------------
    __shared__ float lds[8 * 8];
    int wave = threadIdx.x >> 5;
    if (lane < 8) lds[wave * 8 + lane] = tot;
    __syncthreads();

    if (threadIdx.x < 8) {
        float s = 0.f;
        #pragma unroll
        for (int w = 0; w < 8; ++w) s += lds[w * 8 + threadIdx.x];
        atomicAdd(&ws[threadIdx.x], s);        // global_atomic_add_f32
    }
}

__global__ void pnu_final(const float* __restrict__ ws,
                          const float* __restrict__ p_ratio,
                          float* __restrict__ out) {
    if (threadIdx.x == 0 && blockIdx.x == 0) {
        float spn = ws[0];                      // sum idx_p * sigmoid(-y)
        float spp = ws[1];                      // sum idx_p * sigmoid(y)
        float snp = ws[2];                      // sum idx_n * sigmoid(y)
        float sup = ws[3];                      // sum idx_u * sigmoid(y)
        float np  = fmaxf(1.0f, ws[4]);
        float nn  = fmaxf(1.0f, ws[5]);
        float nu  = fmaxf(1.0f, ws[6]);

        float R_P_plus  = spn / np;
        float R_P_minus = spp / np;
        float R_N_minus = snp / nn;
        float R_U_minus = sup / nu;

        float pr0 = p_ratio[0], pr1 = p_ratio[1];
        float R_PN = pr0 * R_P_plus + pr1 * R_N_minus;
        float R_PU = pr0 * (R_P_plus - R_P_minus) + R_U_minus;

        out[0] = (1.0f - PNU_ETA) * R_PN + PNU_ETA * R_PU;
    }
}

extern "C" void kernel_launch(void* const* d_in, const int* in_sizes, int n_in,
                              void* d_out, int out_size, void* d_ws, size_t ws_size,
                              hipStream_t stream) {
    const float* y       = (const float*)d_in[0];
    const int*   t       = (const int*)d_in[1];
    const float* p_ratio = (const float*)d_in[2];
    float*       ws      = (float*)d_ws;
    float*       out     = (float*)d_out;
    const int n = in_sizes[0];

    pnu_init_ws<<<1, 32, 0, stream>>>(ws);

    int nvec   = n >> 2;
    int blocks = (nvec + 255) / 256;
    if (blocks > 4096) blocks = 4096;
    if (blocks < 1)    blocks = 1;
    pnu_main<<<blocks, 256, 0, stream>>>(y, t, ws, n);

    pnu_final<<<1, 32, 0, stream>>>(ws, p_ratio, out);
}